// MultiHeadDynamicSequenceChunker_47141561041257
// MI455X (gfx1250) — compile-verified
//
#include <hip/hip_runtime.h>

typedef __attribute__((ext_vector_type(2))) float v2f;
typedef __attribute__((ext_vector_type(8))) float v8f;

namespace {
constexpr int   kB   = 8;
constexpr int   kL   = 4096;
constexpr int   kD   = 512;    // token dim == K of GEMM
constexpr int   kNW  = 1024;   // 2*DQK columns of w_qk
constexpr int   kTM  = 128;    // token rows per block (8 waves x 16)
constexpr int   kPAD = 72;     // LDS row stride (floats); K and K+2 rows -> disjoint banks
constexpr float kEPS = 1e-8f;
}

// ---------------------------------------------------------------------------
// Kernel 1: fused  q/k GEMM (WMMA f32 16x16x4)  +  cosine -> probs reduction
// grid = B * (L/128) blocks of 256 threads (8 waves); wave w owns token rows
// [tBase+16w, tBase+16w+16). For each 64-wide N group (8 groups cover DQK=512)
// it accumulates 16x16 tiles of q and k_prev and folds them elementwise into
// per-(row,col) partial sums of q*kp, q*q, kp*kp.
// ---------------------------------------------------------------------------
__global__ __launch_bounds__(256) void qk_probs_kernel(
    const float* __restrict__ tokens,
    const float* __restrict__ w_qk,
    const float* __restrict__ start_key,
    float* __restrict__ probs,
    float* __restrict__ stats)   // [16]: boundary counts [0..7], prob sums [8..15]
{
  extern __shared__ float lds[];           // 2 panels of 512 x kPAD floats
  float* wqL = lds;
  float* wkL = lds + 512 * kPAD;

  const int tid  = threadIdx.x;
  const int wave = tid >> 5;
  const int lane = tid & 31;
  const int half = lane >> 4;              // K-half select (lanes 16-31 -> K+2)
  const int l16  = lane & 15;

  const int blocksPerBatch = kL / kTM;     // 32
  const int b     = blockIdx.x / blocksPerBatch;
  const int tBase = (blockIdx.x % blocksPerBatch) * kTM;

  // A-fragment row for this lane (M = lane%16), within batch
  const int  rowT  = tBase + wave * 16 + l16;
  const long qRow  = (long)b * kL + rowT;
  const long kpRow = (long)b * kL + (rowT > 0 ? rowT - 1 : 0);  // clamped; t==0 fixed below

  const float* Aq  = tokens + qRow  * kD + 2 * half;
  const float* Akp = tokens + kpRow * kD + 2 * half;

  const v8f z8 = {0.f, 0.f, 0.f, 0.f, 0.f, 0.f, 0.f, 0.f};
  v8f accDot = z8, accQQ = z8, accKK = z8;

  const bool t0wave = (tBase == 0) && (wave == 0);  // tile contains t==0 at (vgpr 0, half 0)

  for (int g = 0; g < 8; ++g) {
    // --- cooperatively stage Wq / Wk panels: cols [g*64, g*64+64), all K ---
    for (int i = tid; i < 512 * 16; i += 256) {     // 16 float4 per row
      const int k  = i >> 4;
      const int n4 = (i & 15) << 2;
      const float4 vq = *(const float4*)(w_qk + (long)k * kNW + g * 64 + n4);
      const float4 vk = *(const float4*)(w_qk + (long)k * kNW + 512 + g * 64 + n4);
      *(float4*)(wqL + k * kPAD + n4) = vq;
      *(float4*)(wkL + k * kPAD + n4) = vk;
    }
    __syncthreads();

    v8f qa[4], pa[4];
#pragma unroll
    for (int nt = 0; nt < 4; ++nt) { qa[nt] = z8; pa[nt] = z8; }

#pragma unroll 4
    for (int k = 0; k < kD; k += 4) {
      const v2f aq = *(const v2f*)(Aq + k);
      const v2f ak = *(const v2f*)(Akp + k);
      const int kr = k + 2 * half;
#pragma unroll
      for (int nt = 0; nt < 4; ++nt) {
        v2f bq, bk;
        bq[0] = wqL[kr * kPAD + nt * 16 + l16];
        bq[1] = wqL[(kr + 1) * kPAD + nt * 16 + l16];
        bk[0] = wkL[kr * kPAD + nt * 16 + l16];
        bk[1] = wkL[(kr + 1) * kPAD + nt * 16 + l16];
        qa[nt] = __builtin_amdgcn_wmma_f32_16x16x4_f32(
            false, aq, false, bq, (short)0, qa[nt], false, false);
        pa[nt] = __builtin_amdgcn_wmma_f32_16x16x4_f32(
            false, ak, false, bk, (short)0, pa[nt], false, false);
      }
    }

    // --- fold tiles into per-row stats (elementwise over C layout) ---
#pragma unroll
    for (int nt = 0; nt < 4; ++nt) {
      const float sk = start_key[g * 64 + nt * 16 + l16];
#pragma unroll
      for (int e = 0; e < 8; ++e) {
        const float qv = qa[nt][e];
        float       kv = pa[nt][e];
        if (e == 0 && half == 0 && t0wave) kv = sk;  // row t==0: k_prev = start_key
        accDot[e] += qv * kv;
        accQQ[e]  += qv * qv;
        accKK[e]  += kv * kv;
      }
    }
    __syncthreads();
  }

  // --- reduce each row across the 16 lanes of its half; lanes 0/16 emit ---
  float cntLoc = 0.f, sumLoc = 0.f;
#pragma unroll
  for (int e = 0; e < 8; ++e) {
    float d  = accDot[e];
    float qq = accQQ[e];
    float kk = accKK[e];
#pragma unroll
    for (int m = 8; m >= 1; m >>= 1) {
      d  += __shfl_xor(d,  m, 32);
      qq += __shfl_xor(qq, m, 32);
      kk += __shfl_xor(kk, m, 32);
    }
    if (l16 == 0) {
      const int t  = tBase + wave * 16 + e + 8 * half;
      const float na = fmaxf(sqrtf(qq), kEPS);
      const float nb = fmaxf(sqrtf(kk), kEPS);
      const float cs = d / (na * nb);
      const float pr = (1.0f - cs) * 0.5f;
      probs[(long)b * kL + t] = pr;
      const bool bd = (t == 0) || (pr > 0.5f);
      cntLoc += bd ? 1.0f : 0.0f;
      sumLoc += pr;
    }
  }
  if (l16 == 0) {
    atomicAdd(&stats[b], cntLoc);
    atomicAdd(&stats[8 + b], sumLoc);
  }
}

// ---------------------------------------------------------------------------
// Kernel 2: per-feature sequential chunk recurrence + broadcast to all tokens
// grid = B * (D/64) blocks of 64 threads; each thread owns one feature.
// ---------------------------------------------------------------------------
__global__ __launch_bounds__(64) void scan_kernel(
    const float* __restrict__ tokens,
    const float* __restrict__ probs,
    float* __restrict__ ups)
{
  const int groups = kD / 64;                       // 8
  const int b  = blockIdx.x / groups;
  const int d  = (blockIdx.x % groups) * 64 + threadIdx.x;
  const float* tk = tokens + (long)b * kL * kD + d;
  const float* pr = probs  + (long)b * kL;
  float*       up = ups    + (long)b * kL * kD + d;

  float y = 0.f;
  for (int t = 0; t < kL; ++t) {
    const float p = pr[t];
    const bool bd = (t == 0) || (p > 0.5f);
    if (bd) y = (1.0f - p) * y + tk[(long)t * kD] * p;
    up[(long)t * kD] = y;                           // scale forward-value == 1.0
  }
}

// ---------------------------------------------------------------------------
// Kernel 3: aux ratio loss scalar
// ---------------------------------------------------------------------------
__global__ void aux_kernel(const float* __restrict__ stats,
                           float* __restrict__ outAux)
{
  if (threadIdx.x == 0 && blockIdx.x == 0) {
    float s = 0.f;
    for (int b = 0; b < kB; ++b) {
      const float F = stats[b] * (1.0f / kL);
      const float G = stats[8 + b] * (1.0f / kL);
      // N=6: (N/(N-1)) * ((N-1)*F*G + (1-F)*(1-G))
      s += (6.0f / 5.0f) * (5.0f * F * G + (1.0f - F) * (1.0f - G));
    }
    outAux[0] = (s / (float)kB) * 0.03f;
  }
}

// ---------------------------------------------------------------------------
extern "C" void kernel_launch(void* const* d_in, const int* in_sizes, int n_in,
                              void* d_out, int out_size, void* d_ws, size_t ws_size,
                              hipStream_t stream) {
  (void)in_sizes; (void)n_in; (void)out_size; (void)ws_size;
  const float* tokens = (const float*)d_in[0];   // [8, 4096, 512]
  const float* w_qk   = (const float*)d_in[1];   // [512, 1024]
  const float* skey   = (const float*)d_in[2];   // [1, 512]

  float* ups  = (float*)d_out;                               // 8*4096*512 floats
  float* aux  = ups + (size_t)kB * kL * kD;                  // +1 scalar
  float* probs = (float*)d_ws;                               // 32768 floats
  float* stats = probs + (size_t)kB * kL;                    // 16 floats

  hipMemsetAsync(stats, 0, 16 * sizeof(float), stream);

  const size_t shmem = (size_t)2 * 512 * kPAD * sizeof(float);  // 288 KB (<320 KB/WGP)
  qk_probs_kernel<<<dim3(kB * (kL / kTM)), dim3(256), shmem, stream>>>(
      tokens, w_qk, skey, probs, stats);
  scan_kernel<<<dim3(kB * (kD / 64)), dim3(64), 0, stream>>>(tokens, probs, ups);
  aux_kernel<<<1, 32, 0, stream>>>(stats, aux);
}